// STFT_45165876085022
// MI455X (gfx1250) — compile-verified
//
#include <hip/hip_runtime.h>

typedef __attribute__((ext_vector_type(2))) float v2f;
typedef __attribute__((ext_vector_type(4))) float v4f;
typedef __attribute__((ext_vector_type(8))) float v8f;

#define B_BATCH   32
#define N_IN      160000
#define FILT      800
#define HOP       200
#define PAD       400
#define PADDED_N  (N_IN + 2 * PAD)       // 160800
#define C_FULL    802                    // CUTOFF*2 filters
#define C_PAD     808                    // K padded to multiple of 8
#define T_FRAMES  801
#define MROWS     (B_BATCH * T_FRAMES)   // 25632, multiple of 16
#define MT        (MROWS / 16)           // 1602 M-tiles
#define NT64      13                     // ceil(832/64) covers both 802 and 800
#define WAVES_TOTAL (MT * NT64)          // 20826
#define TINY_F32  1.17549435e-38f

__device__ __forceinline__ int imin(int a, int b) { return a < b ? a : b; }

#define WMMA_F32(A, Bf, Cf) \
    __builtin_amdgcn_wmma_f32_16x16x4_f32(false, (A), false, (Bf), (short)0, (Cf), false, false)

// 8 wmmas from one b128 A-fragment and four b128 B-fragments.
// Lanes 0-15 hold K-base 8c..8c+3, lanes 16-31 hold 8c+4..8c+7:
//   .xy halves form K-chunk {8c,8c+1,8c+4,8c+5}, .zw halves {8c+2,8c+3,8c+6,8c+7}.
#define WMMA_GROUP8(A4, B40, B41, B42, B43)              \
    do {                                                 \
        c0 = WMMA_F32((A4).xy, (B40).xy, c0);            \
        c1 = WMMA_F32((A4).xy, (B41).xy, c1);            \
        c2 = WMMA_F32((A4).xy, (B42).xy, c2);            \
        c3 = WMMA_F32((A4).xy, (B43).xy, c3);            \
        c0 = WMMA_F32((A4).zw, (B40).zw, c0);            \
        c1 = WMMA_F32((A4).zw, (B41).zw, c1);            \
        c2 = WMMA_F32((A4).zw, (B42).zw, c2);            \
        c3 = WMMA_F32((A4).zw, (B43).zw, c3);            \
    } while (0)

// ---------------------------------------------------------------------------
// Materialize reflect-padded input: xp[b, p] = x[b, reflect(p - PAD)]
// ---------------------------------------------------------------------------
__global__ void __launch_bounds__(256)
stft_pad(const float* __restrict__ x, float* __restrict__ xp) {
    const int gid = blockIdx.x * blockDim.x + threadIdx.x;
    if (gid >= B_BATCH * PADDED_N) return;
    const int b = gid / PADDED_N;
    const int p = gid - b * PADDED_N;
    const int q = p - PAD;
    const int idx = (q < 0) ? -q : ((q >= N_IN) ? (2 * (N_IN - 1) - q) : q);
    xp[gid] = x[b * N_IN + idx];
}

// ---------------------------------------------------------------------------
// Transpose inverse_basis into ibT[n, c] with c padded 802 -> 808 (zeros).
// ---------------------------------------------------------------------------
__global__ void __launch_bounds__(256)
stft_transpose_ib(const float* __restrict__ ib, float* __restrict__ ibT) {
    const int gid = blockIdx.x * blockDim.x + threadIdx.x;
    if (gid >= FILT * C_PAD) return;
    const int n = gid / C_PAD;
    const int c = gid - n * C_PAD;
    ibT[gid] = (c < C_FULL) ? ib[c * FILT + n] : 0.0f;
}

// ---------------------------------------------------------------------------
// Forward STFT as GEMM:  ft[row=b*T+t, c] = sum_l xp[b, t*HOP + l] * fb[c, l]
// One wave -> 16(M) x 64(N) tile, K = 800 in groups of 8 (b128 fragments).
// Software-pipelined; prefetched fragments pinned with an asm barrier.
// ---------------------------------------------------------------------------
__global__ void __launch_bounds__(256)
stft_fwd_gemm(const float* __restrict__ xp, const float* __restrict__ fb,
              float* __restrict__ ft) {
    const int wave = (blockIdx.x * blockDim.x + threadIdx.x) >> 5;
    if (wave >= WAVES_TOTAL) return;              // wave-uniform: EXEC stays all-1s
    const int lane = threadIdx.x & 31;
    const int tm = wave % MT;
    const int tn = wave / MT;
    const int m0 = tm * 16;
    const int n0 = tn * 64;

    const int mrow  = m0 + (lane & 15);
    const int b     = mrow / T_FRAMES;
    const int t     = mrow - b * T_FRAMES;
    const int koff4 = (lane >> 4) << 2;           // 0 or 4
    const int nl    = n0 + (lane & 15);

    const float* ap  = xp + (size_t)b * PADDED_N + t * HOP + koff4;
    const float* bp0 = fb + (size_t)imin(nl,      C_FULL - 1) * FILT + koff4;
    const float* bp1 = fb + (size_t)imin(nl + 16, C_FULL - 1) * FILT + koff4;
    const float* bp2 = fb + (size_t)imin(nl + 32, C_FULL - 1) * FILT + koff4;
    const float* bp3 = fb + (size_t)imin(nl + 48, C_FULL - 1) * FILT + koff4;

    v8f c0 = {}, c1 = {}, c2 = {}, c3 = {};

    v4f a  = *(const v4f*)(ap);
    v4f b0 = *(const v4f*)(bp0);
    v4f b1 = *(const v4f*)(bp1);
    v4f b2 = *(const v4f*)(bp2);
    v4f b3 = *(const v4f*)(bp3);

    for (int k = 8; k < FILT; k += 8) {
        v4f an  = *(const v4f*)(ap  + k);
        v4f b0n = *(const v4f*)(bp0 + k);
        v4f b1n = *(const v4f*)(bp1 + k);
        v4f b2n = *(const v4f*)(bp2 + k);
        v4f b3n = *(const v4f*)(bp3 + k);
        // pin prefetched fragments so the coalescer can't collapse the pipeline
        asm volatile("" : "+v"(an), "+v"(b0n), "+v"(b1n), "+v"(b2n), "+v"(b3n));
        WMMA_GROUP8(a, b0, b1, b2, b3);
        a = an; b0 = b0n; b1 = b1n; b2 = b2n; b3 = b3n;
    }
    WMMA_GROUP8(a, b0, b1, b2, b3);

    // C/D layout: VGPR v -> M = v (lanes 0-15) or v+8 (lanes 16-31); N = lane&15
    // Pad columns 802..807 are written as 0 so the inverse GEMM reads clean zeros.
    const int rbase = m0 + ((lane >> 4) ? 8 : 0);
    const int col   = lane & 15;
#pragma unroll
    for (int v = 0; v < 8; ++v) {
        const size_t row = (size_t)(rbase + v) * C_PAD;
        int n;
        n = n0 + col;      if (n < C_PAD) ft[row + n] = (n < C_FULL) ? c0[v] : 0.0f;
        n = n0 + col + 16; if (n < C_PAD) ft[row + n] = (n < C_FULL) ? c1[v] : 0.0f;
        n = n0 + col + 32; if (n < C_PAD) ft[row + n] = (n < C_FULL) ? c2[v] : 0.0f;
        n = n0 + col + 48; if (n < C_PAD) ft[row + n] = (n < C_FULL) ? c3[v] : 0.0f;
    }
}

// ---------------------------------------------------------------------------
// Inverse GEMM:  frames[row, l] = sum_c ft[row, c] * ibT[l, c],  K = 808
// (recombine == ft: mag*cos(atan2(im,re)) = re, mag*sin(...) = im exactly)
// K pad columns are zero in both ft and ibT -> uniform pipelined loop.
// ---------------------------------------------------------------------------
__global__ void __launch_bounds__(256)
stft_inv_gemm(const float* __restrict__ ft, const float* __restrict__ ibT,
              float* __restrict__ frames) {
    const int wave = (blockIdx.x * blockDim.x + threadIdx.x) >> 5;
    if (wave >= WAVES_TOTAL) return;
    const int lane = threadIdx.x & 31;
    const int tm = wave % MT;
    const int tn = wave / MT;
    const int m0 = tm * 16;
    const int n0 = tn * 64;

    const int mrow  = m0 + (lane & 15);
    const int koff4 = (lane >> 4) << 2;
    const int nl    = n0 + (lane & 15);

    const float* ap  = ft  + (size_t)mrow * C_PAD + koff4;
    const float* bp0 = ibT + (size_t)imin(nl,      FILT - 1) * C_PAD + koff4;
    const float* bp1 = ibT + (size_t)imin(nl + 16, FILT - 1) * C_PAD + koff4;
    const float* bp2 = ibT + (size_t)imin(nl + 32, FILT - 1) * C_PAD + koff4;
    const float* bp3 = ibT + (size_t)imin(nl + 48, FILT - 1) * C_PAD + koff4;

    v8f c0 = {}, c1 = {}, c2 = {}, c3 = {};

    v4f a  = *(const v4f*)(ap);
    v4f b0 = *(const v4f*)(bp0);
    v4f b1 = *(const v4f*)(bp1);
    v4f b2 = *(const v4f*)(bp2);
    v4f b3 = *(const v4f*)(bp3);

    for (int k = 8; k < C_PAD; k += 8) {
        v4f an  = *(const v4f*)(ap  + k);
        v4f b0n = *(const v4f*)(bp0 + k);
        v4f b1n = *(const v4f*)(bp1 + k);
        v4f b2n = *(const v4f*)(bp2 + k);
        v4f b3n = *(const v4f*)(bp3 + k);
        asm volatile("" : "+v"(an), "+v"(b0n), "+v"(b1n), "+v"(b2n), "+v"(b3n));
        WMMA_GROUP8(a, b0, b1, b2, b3);
        a = an; b0 = b0n; b1 = b1n; b2 = b2n; b3 = b3n;
    }
    WMMA_GROUP8(a, b0, b1, b2, b3);

    const int rbase = m0 + ((lane >> 4) ? 8 : 0);
    const int col   = lane & 15;
#pragma unroll
    for (int v = 0; v < 8; ++v) {
        const size_t row = (size_t)(rbase + v) * FILT;
        int n;
        n = n0 + col;      if (n < FILT) frames[row + n] = c0[v];
        n = n0 + col + 16; if (n < FILT) frames[row + n] = c1[v];
        n = n0 + col + 32; if (n < FILT) frames[row + n] = c2[v];
        n = n0 + col + 48; if (n < FILT) frames[row + n] = c3[v];
    }
}

// ---------------------------------------------------------------------------
// Overlap-add + analytic window_sumsquare (Hann^2) + scale + crop.
// <= 4 frame contributions per sample; no atomics -> deterministic.
// ---------------------------------------------------------------------------
__global__ void __launch_bounds__(256)
stft_ola(const float* __restrict__ frames, float* __restrict__ out) {
    const int gid = blockIdx.x * blockDim.x + threadIdx.x;
    if (gid >= B_BATCH * N_IN) return;
    const int b = gid / N_IN;
    const int i = gid - b * N_IN;
    const int n = i + PAD;                        // position in out_len (160800) coords

    int tlo = (n - (FILT - 1) + HOP - 1) / HOP;
    if (tlo < 0) tlo = 0;
    int thi = n / HOP;
    if (thi > T_FRAMES - 1) thi = T_FRAMES - 1;

    const float two_pi_over_L = 6.283185307179586f / (float)FILT;
    float acc = 0.0f, ws = 0.0f;
    for (int t = tlo; t <= thi; ++t) {
        const int l = n - t * HOP;                // 0..799
        const float w = 0.5f - 0.5f * cosf(two_pi_over_L * (float)l);
        ws  += w * w;
        acc += frames[(size_t)(b * T_FRAMES + t) * FILT + l];
    }
    const float v = (ws > TINY_F32) ? (acc / ws) : acc;
    out[gid] = v * ((float)FILT / (float)HOP);    // * 4
}

// ---------------------------------------------------------------------------
extern "C" void kernel_launch(void* const* d_in, const int* in_sizes, int n_in,
                              void* d_out, int out_size, void* d_ws, size_t ws_size,
                              hipStream_t stream) {
    const float* x  = (const float*)d_in[0];   // (32, 160000)
    const float* fb = (const float*)d_in[1];   // (802, 800) forward_basis
    const float* ib = (const float*)d_in[2];   // (802, 800) inverse_basis
    float* out = (float*)d_out;                // (32, 1, 160000)

    char* ws = (char*)d_ws;
    float* xp     = (float*)ws;                                  // 32 x 160800
    ws += (size_t)B_BATCH * PADDED_N * sizeof(float);            // 20,582,400 B
    float* ibT    = (float*)ws;                                  // 800 x 808
    ws += (size_t)FILT * C_PAD * sizeof(float);                  //  2,585,600 B
    float* ft     = (float*)ws;                                  // 25632 x 808
    ws += (size_t)MROWS * C_PAD * sizeof(float);                 // 82,842,624 B
    float* frames = (float*)ws;                                  // 25632 x 800

    const int pad_total = B_BATCH * PADDED_N;
    stft_pad<<<(pad_total + 255) / 256, 256, 0, stream>>>(x, xp);

    const int tr_total = FILT * C_PAD;
    stft_transpose_ib<<<(tr_total + 255) / 256, 256, 0, stream>>>(ib, ibT);

    const int gemm_blocks = (WAVES_TOTAL + 7) / 8;               // 8 waves / block
    stft_fwd_gemm<<<gemm_blocks, 256, 0, stream>>>(xp, fb, ft);
    stft_inv_gemm<<<gemm_blocks, 256, 0, stream>>>(ft, ibT, frames);

    const int ola_total = B_BATCH * N_IN;
    stft_ola<<<(ola_total + 255) / 256, 256, 0, stream>>>(frames, out);
}